// TransposeCASSI_34007551050255
// MI455X (gfx1250) — compile-verified
//
#include <hip/hip_runtime.h>

#define BB   8
#define MD   256
#define LW   31
#define N2   286                 // Md + L - 1; max(m+l)=285 so the %N2 is a no-op
#define PERBI (MD * LW)          // 7936 core elements per (b,i)
#define NBI  (BB * MD)           // 2048 (b,i) pairs
#define OW   (MD * 2)            // upsampled height/width = 512

// ---- CDNA5 async staging: copy the (b,i) Y row (286 x float2 = 2288 B) into LDS ----
// global_load_async_to_lds_b64: VDST = LDS byte-offset VGPR, VADDR = 32-bit global
// offset VGPR (GVS mode), SADDR = uniform 64-bit SGPR base. Tracked by ASYNCcnt.
__device__ __forceinline__ void stage_Y_row_async(float2* sY,
                                                  const float2* __restrict__ Y2,
                                                  int bi) {
    const int tid = threadIdx.x;
    const unsigned long long base = (unsigned long long)(Y2 + (size_t)bi * N2);

    // flat shared-pointer low 32 bits == LDS byte offset (LDS aperture, ISA 10.2)
    unsigned lds0  = (unsigned)(uintptr_t)(&sY[tid]);
    unsigned goff0 = (unsigned)tid * 8u;
    asm volatile("global_load_async_to_lds_b64 %0, %1, %2"
                 :: "v"(lds0), "v"(goff0), "s"(base) : "memory");

    if (tid < N2 - 256) {   // remaining 30 float2's
        unsigned lds1  = (unsigned)(uintptr_t)(&sY[256 + tid]);
        unsigned goff1 = (256u + (unsigned)tid) * 8u;
        asm volatile("global_load_async_to_lds_b64 %0, %1, %2"
                     :: "v"(lds1), "v"(goff1), "s"(base) : "memory");
    }

    asm volatile("s_wait_asynccnt 0" ::: "memory");   // my wave's copies landed in LDS
    __syncthreads();                                  // all waves' copies visible
}

__global__ void zero_max_kernel(unsigned* g) { g[0] = 0u; }

// ---- Pass 1: global max of X[b,i,m,l] = sum_s H[b,i,m,l,s]*Y[b,i,m+l,s] ----
__global__ __launch_bounds__(256)
void max_kernel(const float* __restrict__ Y, const float* __restrict__ H,
                unsigned* __restrict__ gmax) {
    __shared__ float2 sY[N2];
    const float2* Y2 = (const float2*)Y;
    const int bi = blockIdx.x;                         // b*MD + i
    stage_Y_row_async(sY, Y2, bi);

    const float2* __restrict__ Hrow = (const float2*)H + (size_t)bi * PERBI;

    float vmax = 0.0f;
    for (int it = 0; it < LW; ++it) {                  // 31*256 == PERBI exactly
        int e = it * 256 + threadIdx.x;
        __builtin_prefetch(&Hrow[e + 2048], 0, 0);     // global_prefetch_b8, ~8 iters ahead
        int l = e % LW, m = e / LW;
        float2 h = Hrow[e];                            // stride-1 b64, fully coalesced
        float2 y = sY[m + l];                          // ds_load_b64 from staged row
        vmax = fmaxf(vmax, h.x * y.x + h.y * y.y);
    }

    #pragma unroll
    for (int off = 16; off > 0; off >>= 1)             // wave32 reduction
        vmax = fmaxf(vmax, __shfl_xor(vmax, off, 32));

    __shared__ unsigned bmax;
    if (threadIdx.x == 0) bmax = 0u;
    __syncthreads();
    // X >= 0 always (sums of products of uniforms in [0,1)), so uint-bit max == float max
    if ((threadIdx.x & 31) == 0) atomicMax(&bmax, __float_as_uint(vmax));
    __syncthreads();
    if (threadIdx.x == 0) atomicMax(gmax, bmax);
}

// ---- Pass 2: recompute (H is L2-resident now), divide by max, 2x2 upsample ----
__global__ __launch_bounds__(256)
void scale_upsample_kernel(const float* __restrict__ Y, const float* __restrict__ H,
                           const unsigned* __restrict__ gmax,
                           float* __restrict__ out) {
    __shared__ float2 sY[N2];
    const float2* Y2 = (const float2*)Y;
    const int bi = blockIdx.x;
    stage_Y_row_async(sY, Y2, bi);

    const float2* __restrict__ Hrow = (const float2*)H + (size_t)bi * PERBI;
    const float maxv = __uint_as_float(*gmax);         // uniform scalar load

    const int i = bi % MD;
    const int b = bi / MD;
    const size_t rpitch = (size_t)OW * LW;             // one upsampled row

    for (int it = 0; it < LW; ++it) {
        int e = it * 256 + threadIdx.x;
        __builtin_prefetch(&Hrow[e + 2048], 0, 0);
        int l = e % LW, m = e / LW;
        float2 h = Hrow[e];
        float2 y = sY[m + l];
        float v = (h.x * y.x + h.y * y.y) / maxv;      // IEEE divide, matches reference

        // out[b][2i+di][2m+dm][l], shape (8, 512, 512, 31); write-once 260 MB
        size_t row0 = ((size_t)(b * OW + 2 * i) * OW + 2 * m) * LW + l;
        __builtin_nontemporal_store(v, &out[row0]);
        __builtin_nontemporal_store(v, &out[row0 + LW]);
        __builtin_nontemporal_store(v, &out[row0 + rpitch]);
        __builtin_nontemporal_store(v, &out[row0 + rpitch + LW]);
    }
}

extern "C" void kernel_launch(void* const* d_in, const int* in_sizes, int n_in,
                              void* d_out, int out_size, void* d_ws, size_t ws_size,
                              hipStream_t stream) {
    const float* Y = (const float*)d_in[0];   // (8, 256, 286, 2) f32
    const float* H = (const float*)d_in[1];   // (8, 256, 256, 31, 2) f32
    float* out     = (float*)d_out;           // (8, 512, 512, 31) f32
    unsigned* gmax = (unsigned*)d_ws;         // 4 B scratch; must re-zero every launch

    zero_max_kernel<<<1, 1, 0, stream>>>(gmax);
    max_kernel<<<NBI, 256, 0, stream>>>(Y, H, gmax);
    scale_upsample_kernel<<<NBI, 256, 0, stream>>>(Y, H, gmax, out);
}